// DeepGMM_59330678227290
// MI455X (gfx1250) — compile-verified
//
#include <hip/hip_runtime.h>
#include <math.h>

#define NB 256
#define SS 10
#define KK 32
#define DD 64
#define NSAMP (NB * SS)

#define REGC 1.0f
#define JITC 0.1f
#define LOG2PI_F 1.8378770664093453f

typedef __attribute__((ext_vector_type(2))) float v2f;
typedef __attribute__((ext_vector_type(8))) float v8f;

// f32 WMMA (16x16x4) — guard so the host pass (no amdgcn builtins) still parses,
// and so a missing builtin degrades to a geometry-identical VALU path.
#if defined(__HIP_DEVICE_COMPILE__) && __has_builtin(__builtin_amdgcn_wmma_f32_16x16x4_f32)
#define USE_WMMA_F32 1
#else
#define USE_WMMA_F32 0
#endif

__device__ __forceinline__ float warpSum(float v) {
#pragma unroll
  for (int m = 16; m >= 1; m >>= 1) v += __shfl_xor(v, m, 32);
  return v;
}
__device__ __forceinline__ float warpMax(float v) {
#pragma unroll
  for (int m = 16; m >= 1; m >>= 1) v = fmaxf(v, __shfl_xor(v, m, 32));
  return v;
}
__device__ __forceinline__ float softplusf(float x) {
  return fmaxf(x, 0.0f) + log1pf(expf(-fabsf(x)));
}
__device__ __forceinline__ float blockSum64(float v, float* part, int lane, int wid) {
  v = warpSum(v);
  if (lane == 0) part[wid] = v;
  __syncthreads();
  float r = part[0] + part[1];
  __syncthreads();
  return r;
}

// ---------------------------------------------------------------------------
// K0: per-cluster Sherman-Morrison precompute for Sigma_phi
//   a = 1/(phi_diags+REG), u = a*f, c = 1/(1+f.u), r = inv(Sigma_phi)@phi_mu
// ---------------------------------------------------------------------------
__global__ void __launch_bounds__(KK) k_clusters(
    const float* __restrict__ phi_diags, const float* __restrict__ phi_factors,
    const float* __restrict__ phi_mus, float* __restrict__ aK,
    float* __restrict__ uK, float* __restrict__ rK, float* __restrict__ cK) {
  const int k = threadIdx.x;
  float s1 = 0.f, umu = 0.f;
  for (int d = 0; d < DD; ++d) {
    float pd = phi_diags[k * DD + d] + REGC;
    float ad = 1.f / pd;
    float fd = phi_factors[k * DD + d];
    float ud = ad * fd;
    s1 += fd * ud;
    umu += ud * phi_mus[k * DD + d];
  }
  float c = 1.f / (1.f + s1);
  cK[k] = c;
  for (int d = 0; d < DD; ++d) {
    float pd = phi_diags[k * DD + d] + REGC;
    float ad = 1.f / pd;
    float fd = phi_factors[k * DD + d];
    float ud = ad * fd;
    aK[k * DD + d] = ad;
    uK[k * DD + d] = ud;
    rK[k * DD + d] = ad * phi_mus[k * DD + d] - c * ud * umu;
  }
}

// ---------------------------------------------------------------------------
// K1/K4: dual affine GEMM  out0 = A@W0+b0 ; out1 = softplus(A@W1+b1)+JIT
// One wave per 16x16 output tile; WMMA f32 16x16x4 over the K=64 dimension.
// grid.x = (M/16)*4*2 blocks of 32 threads.
// ---------------------------------------------------------------------------
__global__ void __launch_bounds__(32) k_gemm2(
    const float* __restrict__ A, int M,
    const float* __restrict__ W0, const float* __restrict__ b0, float* __restrict__ out0,
    const float* __restrict__ W1, const float* __restrict__ b1, float* __restrict__ out1) {
  const int Mtiles = M >> 4;
  int tile = blockIdx.x;
  const int which = tile / (Mtiles * 4);
  tile -= which * Mtiles * 4;
  const int M0 = (tile >> 2) << 4;
  const int N0 = (tile & 3) << 4;
  const int lane = threadIdx.x;
  const int lm = lane & 15;
  const int rbase = (lane < 16) ? 0 : 8;
  const float* W = which ? W1 : W0;
  const float* b = which ? b1 : b0;
  float res[8];
#if USE_WMMA_F32
  const int koff = (lane < 16) ? 0 : 2;
  v8f cacc = {};
  for (int k0 = 0; k0 < DD; k0 += 4) {
    v2f av, bv;
    // A frag (16x4): lanes 0-15 hold K={k0,k0+1}, lanes 16-31 K={k0+2,k0+3}
    av.x = A[(size_t)(M0 + lm) * DD + k0 + koff];
    av.y = A[(size_t)(M0 + lm) * DD + k0 + koff + 1];
    // B frag (4x16), row-major W[k][n]
    bv.x = W[(k0 + koff) * DD + N0 + lm];
    bv.y = W[(k0 + koff + 1) * DD + N0 + lm];
    cacc = __builtin_amdgcn_wmma_f32_16x16x4_f32(false, av, false, bv,
                                                 (short)0, cacc, false, false);
  }
#pragma unroll
  for (int r = 0; r < 8; ++r) res[r] = cacc[r];
#else
#pragma unroll
  for (int r = 0; r < 8; ++r) {
    int row = M0 + rbase + r;
    float s = 0.f;
    for (int k = 0; k < DD; ++k) s += A[(size_t)row * DD + k] * W[k * DD + N0 + lm];
    res[r] = s;
  }
#endif
#pragma unroll
  for (int r = 0; r < 8; ++r) {
    int row = M0 + rbase + r;
    int col = N0 + lm;
    float v = res[r] + b[col];
    if (which == 0) out0[(size_t)row * DD + col] = v;
    else            out1[(size_t)row * DD + col] = softplusf(v) + JITC;
  }
}

// ---------------------------------------------------------------------------
// K2: z_logits via Sherman-Morrison on diag+rank1 cov, posterior Sigma_tilde
//     (diag(b)+g v v^T) and mu_tilde; plus logsumexp(z_logits) accumulation.
// One wave per (n,k); block = (32 lanes, 8 k-waves), grid = N.
// ---------------------------------------------------------------------------
__global__ void __launch_bounds__(256) k_posterior(
    const float* __restrict__ mu_e, const float* __restrict__ sdiag,
    const float* __restrict__ phi_mus, const float* __restrict__ phi_diags,
    const float* __restrict__ phi_factors, const float* __restrict__ phi_pis,
    const float* __restrict__ aK, const float* __restrict__ uK,
    const float* __restrict__ rK, const float* __restrict__ cK,
    float* __restrict__ zlog, float* __restrict__ bnk, float* __restrict__ vnk,
    float* __restrict__ gnk, float* __restrict__ mut, float* __restrict__ acc) {
  const int n = blockIdx.x;
  const int lane = threadIdx.x;
  const int d0 = lane, d1 = lane + 32;
  __shared__ float zl[KK];

  const float mu0 = mu_e[n * DD + d0], mu1 = mu_e[n * DD + d1];
  const float sd0 = sdiag[n * DD + d0], sd1 = sdiag[n * DD + d1];
  const float ie0 = 1.f / sd0, ie1 = 1.f / sd1;

  for (int k = threadIdx.y; k < KK; k += 8) {
    float pd0 = phi_diags[k * DD + d0] + REGC, pd1 = phi_diags[k * DD + d1] + REGC;
    float f0 = phi_factors[k * DD + d0], f1 = phi_factors[k * DD + d1];
    float dk0 = sd0 + pd0, dk1 = sd1 + pd1;
    float id0 = 1.f / dk0, id1 = 1.f / dk1;
    float df0 = mu0 - phi_mus[k * DD + d0], df1 = mu1 - phi_mus[k * DD + d1];
    float q1 = warpSum(df0 * df0 * id0 + df1 * df1 * id1);
    float q2 = warpSum(df0 * f0 * id0 + df1 * f1 * id1);
    float q3 = warpSum(f0 * f0 * id0 + f1 * f1 * id1);
    float ld = warpSum(logf(dk0) + logf(dk1));
    float Mh = q1 - q2 * q2 / (1.f + q3);
    float lp = -0.5f * (DD * LOG2PI_F + Mh) - 0.5f * (ld + log1pf(q3));
    if (lane == 0) zl[k] = logf(phi_pis[k]) + lp;

    // Sigma_tilde = (diag(inv_e + a) - c u u^T)^{-1} = diag(b) + g v v^T
    float a0 = aK[k * DD + d0], a1 = aK[k * DD + d1];
    float u0 = uK[k * DD + d0], u1 = uK[k * DD + d1];
    float b0 = 1.f / (ie0 + a0), b1 = 1.f / (ie1 + a1);
    float v0 = b0 * u0, v1 = b1 * u1;
    float uv = warpSum(u0 * v0 + u1 * v1);
    float c = cK[k];
    float g = c / (1.f - c * uv);
    float rh0 = mu0 * ie0 + rK[k * DD + d0], rh1 = mu1 * ie1 + rK[k * DD + d1];
    float vr = warpSum(v0 * rh0 + v1 * rh1);
    size_t base = ((size_t)n * KK + k) * DD;
    bnk[base + d0] = b0; bnk[base + d1] = b1;
    vnk[base + d0] = v0; vnk[base + d1] = v1;
    mut[base + d0] = b0 * rh0 + g * v0 * vr;
    mut[base + d1] = b1 * rh1 + g * v1 * vr;
    if (lane == 0) gnk[n * KK + k] = g;
  }
  __syncthreads();
  if (threadIdx.y == 0) {
    float v = zl[lane];
    zlog[n * KK + lane] = v;
    float mx = warpMax(v);
    float se = warpSum(expf(v - mx));
    if (lane == 0) atomicAdd(&acc[1], mx + logf(se));
  }
}

// ---------------------------------------------------------------------------
// Woodbury log-prob for M = diag(dd) + sum_k z_k f_k f_k^T  (rank-32)
// All 64 threads must call. Returns logp (valid on all threads).
// ---------------------------------------------------------------------------
__device__ float lowrank_mix_logp(const float* __restrict__ factors,
                                  const float* zsh, const float* dd,
                                  const float* mmu, const float* xv,
                                  float* Us /*[KK][DD+1]*/, float* Cm /*[KK][KK+1]*/,
                                  float* pv, float* ed, float* part, int tid) {
  const int lane = tid & 31;
  const int wid = tid >> 5;
  ed[tid] = (xv[tid] - mmu[tid]) * rsqrtf(dd[tid]);
  for (int idx = tid; idx < KK * DD; idx += 64) {
    int k = idx >> 6, d = idx & 63;
    Us[k * (DD + 1) + d] = factors[k * DD + d] * sqrtf(zsh[k]) * rsqrtf(dd[d]);
  }
  __syncthreads();
  if (tid < KK) {
    float s = 0.f;
    for (int d = 0; d < DD; ++d) s += Us[tid * (DD + 1) + d] * ed[d];
    pv[tid] = s;
  }
  for (int idx = tid; idx < KK * KK; idx += 64) {
    int i = idx >> 5, j = idx & 31;
    float s = (i == j) ? 1.f : 0.f;
    for (int d = 0; d < DD; ++d) s += Us[i * (DD + 1) + d] * Us[j * (DD + 1) + d];
    Cm[i * (KK + 1) + j] = s;
  }
  __syncthreads();
  // Cholesky of C (32x32), row-owner, cooperative
  for (int kc = 0; kc < KK; ++kc) {
    float ckk = Cm[kc * (KK + 1) + kc];
    __syncthreads();
    float dg = sqrtf(ckk);
    if (tid == kc) Cm[kc * (KK + 1) + kc] = dg;
    if (tid > kc && tid < KK) Cm[tid * (KK + 1) + kc] /= dg;
    __syncthreads();
    if (tid > kc && tid < KK) {
      float l = Cm[tid * (KK + 1) + kc];
      for (int i = kc + 1; i <= tid; ++i)
        Cm[tid * (KK + 1) + i] -= l * Cm[i * (KK + 1) + kc];
    }
    __syncthreads();
  }
  // forward solve L y = p in place (only ||L^{-1}p||^2 is needed)
  for (int kc = 0; kc < KK; ++kc) {
    if (tid == kc) pv[kc] /= Cm[kc * (KK + 1) + kc];
    __syncthreads();
    if (tid > kc && tid < KK) pv[tid] -= Cm[tid * (KK + 1) + kc] * pv[kc];
    __syncthreads();
  }
  float q = ed[tid] * ed[tid] + logf(dd[tid]);
  if (tid < KK) q += 2.f * logf(Cm[tid * (KK + 1) + tid]) - pv[tid] * pv[tid];
  float tot = blockSum64(q, part, lane, wid);
  return -0.5f * (DD * LOG2PI_F + tot);
}

// ---------------------------------------------------------------------------
// K3: per (n,s) sample: softmax z, St = diag + V~^T V~ (WMMA SYRK), 64x64
// Cholesky, x = mt + L eps, and all non-decoder objective terms.
// grid = N*S blocks of 64 threads (2 full waves -> EXEC all ones for WMMA).
// ---------------------------------------------------------------------------
__global__ void __launch_bounds__(64) k_sample(
    const float* __restrict__ zlog, const float* __restrict__ gum,
    const float* __restrict__ gnk, const float* __restrict__ bnk,
    const float* __restrict__ vnk, const float* __restrict__ mut,
    const float* __restrict__ theta_mus, const float* __restrict__ theta_diags,
    const float* __restrict__ theta_factors, const float* __restrict__ theta_pis,
    const float* __restrict__ phi_mus, const float* __restrict__ phi_diags,
    const float* __restrict__ phi_factors,
    const float* __restrict__ mu_e, const float* __restrict__ sdiag,
    const float* __restrict__ eps, float* __restrict__ xout, float* __restrict__ acc) {
  const int ns = blockIdx.x;
  const int n = ns / SS;
  const int tid = threadIdx.x;
  const int lane = tid & 31;
  const int wid = tid >> 5;

  __shared__ float zsh[KK], zlsh[KK], swsh[KK];
  __shared__ float mt[DD], dacc[DD], ddth[DD], ddph[DD], thmu[DD], phmu[DD];
  __shared__ float xl[DD], el[DD], edsh[DD];
  __shared__ float Vl[KK][DD + 1];
  __shared__ float St[DD][DD + 1];
  __shared__ float Cmat[KK][KK + 1];
  __shared__ float pv[KK];
  __shared__ float part[2];

  // softmax((z_logits + gumbel)/TEMP), TEMP=1
  if (wid == 0) {
    float zl = zlog[n * KK + lane];
    float gu = gum[(size_t)ns * KK + lane];
    float lg = zl - logf(-logf(gu));
    float mx = warpMax(lg);
    float e = expf(lg - mx);
    float s = warpSum(e);
    float zk = e / s;
    zsh[lane] = zk;
    zlsh[lane] = zl;
    swsh[lane] = sqrtf(zk * gnk[n * KK + lane]);  // g>0 always (SPD)
  }
  __syncthreads();

  // mixture diagonals / means; stage V~ = sqrt(z*g) * v_{nk}
  {
    const int d = tid;
    float m0 = 0, m1 = 0, m2 = 0, m3 = 0, m4 = 0, m5 = 0;
    for (int k = 0; k < KK; ++k) {
      float zk = zsh[k];
      size_t base = ((size_t)n * KK + k) * DD + d;
      m0 += zk * mut[base];
      m1 += zk * bnk[base];
      m2 += zk * (theta_diags[k * DD + d] + REGC);
      m3 += zk * (phi_diags[k * DD + d] + REGC);
      m4 += zk * theta_mus[k * DD + d];
      m5 += zk * phi_mus[k * DD + d];
    }
    mt[d] = m0; dacc[d] = m1; ddth[d] = m2; ddph[d] = m3; thmu[d] = m4; phmu[d] = m5;
    el[d] = eps[(size_t)ns * DD + d];
    for (int idx = tid; idx < KK * DD; idx += 64) {
      int k = idx >> 6, dd2 = idx & 63;
      Vl[k][dd2] = vnk[((size_t)n * KK + k) * DD + dd2] * swsh[k];
    }
  }
  __syncthreads();

  // St = diag(dacc) + V~^T V~   (64x64x32 SYRK)
#if USE_WMMA_F32
  {
    const int lm = lane & 15;
    const int koff = (lane < 16) ? 0 : 2;
    const int rbase = (lane < 16) ? 0 : 8;
    for (int t = wid * 8; t < wid * 8 + 8; ++t) {
      const int M0 = (t >> 2) << 4;
      const int N0 = (t & 3) << 4;
      v8f cacc = {};
      for (int k0 = 0; k0 < KK; k0 += 4) {
        v2f av, bv;
        av.x = Vl[k0 + koff][M0 + lm];
        av.y = Vl[k0 + koff + 1][M0 + lm];
        bv.x = Vl[k0 + koff][N0 + lm];
        bv.y = Vl[k0 + koff + 1][N0 + lm];
        cacc = __builtin_amdgcn_wmma_f32_16x16x4_f32(false, av, false, bv,
                                                     (short)0, cacc, false, false);
      }
#pragma unroll
      for (int r = 0; r < 8; ++r) {
        int row = M0 + rbase + r, col = N0 + lm;
        float v = cacc[r];
        if (row == col) v += dacc[row];
        St[row][col] = v;
      }
    }
  }
#else
  {
    const int j = tid;
    for (int i = 0; i < DD; ++i) {
      float s = (i == j) ? dacc[j] : 0.f;
      for (int k = 0; k < KK; ++k) s += Vl[k][j] * Vl[k][i];
      St[j][i] = s;
    }
  }
#endif
  __syncthreads();

  // Cholesky of St (64x64), row-owner right-looking
  for (int kc = 0; kc < DD; ++kc) {
    float skk = St[kc][kc];
    __syncthreads();
    float dg = sqrtf(skk);
    if (tid == kc) St[kc][kc] = dg;
    if (tid > kc) St[tid][kc] /= dg;
    __syncthreads();
    if (tid > kc) {
      float l = St[tid][kc];
      for (int i = kc + 1; i <= tid; ++i) St[tid][i] -= l * St[i][kc];
    }
    __syncthreads();
  }

  // x = mt + L @ eps
  {
    float s = mt[tid];
    for (int c = 0; c <= tid; ++c) s += St[tid][c] * el[c];
    xl[tid] = s;
    xout[(size_t)ns * DD + tid] = s;
  }
  __syncthreads();

  float obj0 = 0.f;  // accumulated on tid 0

  // - logp(x; mu_e, diag(sdiag))
  {
    float df = xl[tid] - mu_e[n * DD + tid];
    float sd = sdiag[n * DD + tid];
    float tot = blockSum64(df * df / sd + logf(sd), part, lane, wid);
    if (tid == 0) obj0 -= -0.5f * (DD * LOG2PI_F + tot);
  }
  // + logp(x; theta mixture)    (reuses Vl as Us scratch)
  {
    float lp = lowrank_mix_logp(theta_factors, zsh, ddth, thmu, xl,
                                &Vl[0][0], &Cmat[0][0], pv, edsh, part, tid);
    if (tid == 0) obj0 += lp;
  }
  __syncthreads();
  // - logp(x; phi mixture)
  {
    float lp = lowrank_mix_logp(phi_factors, zsh, ddph, phmu, xl,
                                &Vl[0][0], &Cmat[0][0], pv, edsh, part, tid);
    if (tid == 0) obj0 -= lp;
  }
  __syncthreads();
  // + sum z log theta_pi - sum z_logits * z
  if (wid == 0) {
    float q = zsh[lane] * logf(theta_pis[lane]) - zlsh[lane] * zsh[lane];
    q = warpSum(q);
    if (lane == 0) obj0 += q;
  }
  if (tid == 0) atomicAdd(&acc[0], obj0);
}

// ---------------------------------------------------------------------------
// K5: decoder diagonal-cov log-prob term (per sample)
// ---------------------------------------------------------------------------
__global__ void __launch_bounds__(64) k_decobj(const float* __restrict__ Y,
                                               const float* __restrict__ muy,
                                               const float* __restrict__ sgy,
                                               float* __restrict__ acc) {
  const int ns = blockIdx.x;
  const int n = ns / SS;
  const int tid = threadIdx.x;
  __shared__ float part[2];
  float df = Y[n * DD + tid] - muy[(size_t)ns * DD + tid];
  float sd = sgy[(size_t)ns * DD + tid];
  float v = warpSum(df * df / sd + logf(sd));
  if ((tid & 31) == 0) part[tid >> 5] = v;
  __syncthreads();
  if (tid == 0) atomicAdd(&acc[0], -0.5f * (DD * LOG2PI_F + part[0] + part[1]));
}

__global__ void k_finalize(const float* __restrict__ acc, float* __restrict__ out) {
  out[0] = acc[0] / (float)SS + acc[1];
}

// ---------------------------------------------------------------------------
extern "C" void kernel_launch(void* const* d_in, const int* in_sizes, int n_in,
                              void* d_out, int out_size, void* d_ws, size_t ws_size,
                              hipStream_t stream) {
  (void)in_sizes; (void)n_in; (void)out_size; (void)ws_size;
  const float* Y            = (const float*)d_in[0];
  const float* W_enc_mu     = (const float*)d_in[1];
  const float* b_enc_mu     = (const float*)d_in[2];
  const float* W_enc_sig    = (const float*)d_in[3];
  const float* b_enc_sig    = (const float*)d_in[4];
  const float* W_dec_mu     = (const float*)d_in[5];
  const float* b_dec_mu     = (const float*)d_in[6];
  const float* W_dec_sig    = (const float*)d_in[7];
  const float* b_dec_sig    = (const float*)d_in[8];
  const float* phi_mus      = (const float*)d_in[9];
  const float* phi_diags    = (const float*)d_in[10];
  const float* phi_factors  = (const float*)d_in[11];
  const float* phi_pis      = (const float*)d_in[12];
  const float* theta_mus    = (const float*)d_in[13];
  const float* theta_diags  = (const float*)d_in[14];
  const float* theta_factors= (const float*)d_in[15];
  const float* theta_pis    = (const float*)d_in[16];
  const float* gumbel_u     = (const float*)d_in[17];
  const float* eps          = (const float*)d_in[18];
  float* out = (float*)d_out;

  // workspace layout (~8.5 MB of fp32)
  float* p = (float*)d_ws;
  float* mu_e  = p; p += NB * DD;
  float* sdiag = p; p += NB * DD;
  float* aK    = p; p += KK * DD;
  float* uK    = p; p += KK * DD;
  float* rK    = p; p += KK * DD;
  float* cKv   = p; p += KK;
  float* zlog  = p; p += NB * KK;
  float* gnk   = p; p += NB * KK;
  float* bnk   = p; p += (size_t)NB * KK * DD;
  float* vnk   = p; p += (size_t)NB * KK * DD;
  float* mut   = p; p += (size_t)NB * KK * DD;
  float* xbuf  = p; p += (size_t)NSAMP * DD;
  float* muy   = p; p += (size_t)NSAMP * DD;
  float* sgy   = p; p += (size_t)NSAMP * DD;
  float* acc   = p; p += 4;

  hipMemsetAsync(acc, 0, 4 * sizeof(float), stream);

  k_clusters<<<1, KK, 0, stream>>>(phi_diags, phi_factors, phi_mus, aK, uK, rK, cKv);
  k_gemm2<<<(NB / 16) * 4 * 2, 32, 0, stream>>>(Y, NB, W_enc_mu, b_enc_mu, mu_e,
                                                W_enc_sig, b_enc_sig, sdiag);
  k_posterior<<<NB, dim3(32, 8), 0, stream>>>(mu_e, sdiag, phi_mus, phi_diags,
                                              phi_factors, phi_pis, aK, uK, rK, cKv,
                                              zlog, bnk, vnk, gnk, mut, acc);
  k_sample<<<NSAMP, 64, 0, stream>>>(zlog, gumbel_u, gnk, bnk, vnk, mut,
                                     theta_mus, theta_diags, theta_factors, theta_pis,
                                     phi_mus, phi_diags, phi_factors,
                                     mu_e, sdiag, eps, xbuf, acc);
  k_gemm2<<<(NSAMP / 16) * 4 * 2, 32, 0, stream>>>(xbuf, NSAMP, W_dec_mu, b_dec_mu, muy,
                                                   W_dec_sig, b_dec_sig, sgy);
  k_decobj<<<NSAMP, 64, 0, stream>>>(Y, muy, sgy, acc);
  k_finalize<<<1, 1, 0, stream>>>(acc, out);
}